// Loss_64295660421321
// MI455X (gfx1250) — compile-verified
//
#include <hip/hip_runtime.h>

// MI455X / gfx1250, wave32. Fused loss:
//  contrastive: streaming-tile sim=(h h^T)/T via V_WMMA_F32_16X16X4_F32, online
//               logsumexp. B tiles staged to LDS once per 8-wave workgroup with
//               the Tensor Data Mover (double-buffered, s_wait_tensorcnt).
//  graph:       single HBM pass over G (128MB) computing r_i, G[i,:].a and the
//               (G@all_x).sub_x GEMM; 4 waves/block share all_x addresses (WGP$).

typedef __attribute__((ext_vector_type(2))) float v2f;
typedef __attribute__((ext_vector_type(8))) float v8f;
typedef __attribute__((ext_vector_type(4))) unsigned int u32x4;
typedef __attribute__((ext_vector_type(8))) int i32x8;
typedef __attribute__((ext_vector_type(4))) int i32x4;

#define B_ROWS 4096
#define N_TOT  8192
#define D_DIM  128
#define K_DIM  8192
#define INV_T  2.0f         // 1 / temperature(0.5)
#define LDS_STRIDE 130      // 128 + 2 dword pad per row (bank-conflict-free b64)

#if defined(__has_builtin)
#if __has_builtin(__builtin_amdgcn_tensor_load_to_lds) && \
    __has_builtin(__builtin_amdgcn_s_wait_tensorcnt)
#define USE_TDM 1
#endif
#endif
#ifndef USE_TDM
#define USE_TDM 0
#endif

__device__ __forceinline__ float wave_sum32(float v) {
  #pragma unroll
  for (int off = 16; off; off >>= 1) v += __shfl_xor(v, off, 32);
  return v;
}

// Stage one 16x128 f32 tile (row-major, contiguous in global) into LDS with
// per-row pad of 2 dwords. TDM path: wave 0 issues one tensor_load_to_lds.
__device__ __forceinline__ void stage_issue(float* dstLds, const float* src) {
#if USE_TDM
  if (threadIdx.x < 32) {
    unsigned long long ga = (unsigned long long)(uintptr_t)src;
    unsigned int lds = (unsigned int)(uintptr_t)dstLds;  // low 32b = LDS offset
    u32x4 g0;
    g0[0] = 1u;                                          // count=1 (user D#)
    g0[1] = lds;                                         // lds_addr
    g0[2] = (unsigned int)(ga & 0xFFFFFFFFu);            // global_addr[31:0]
    g0[3] = (unsigned int)((ga >> 32) & 0x1FFFFFFu)      // global_addr[56:32]
          | (2u << 30);                                  // type=2 (image)
    i32x8 g1;
    g1[0] = (int)((2u << 16)    // data_size = 4B
          | (1u << 20)          // pad_enable
          | (6u << 22)          // pad_interval: every 128 dwords
          | (1u << 25));        // pad_amount: 2 dwords
    g1[1] = (int)(128u << 16);  // tensor_dim0 = 128
    g1[2] = (int)(16u << 16);   // tensor_dim1 = 16
    g1[3] = (int)(128u << 16);  // tile_dim0   = 128
    g1[4] = 16;                 // tile_dim1   = 16
    g1[5] = 128;                // tensor_dim0_stride = 128
    g1[6] = 0;
    g1[7] = 0;
    i32x4 z4 = {0, 0, 0, 0};
#if defined(__clang_major__) && (__clang_major__ >= 23)
    i32x8 z8 = {0, 0, 0, 0, 0, 0, 0, 0};
    __builtin_amdgcn_tensor_load_to_lds(g0, g1, z4, z4, z8, 0);
#else
    __builtin_amdgcn_tensor_load_to_lds(g0, g1, z4, z4, 0);
#endif
  }
#else
  for (int e = threadIdx.x; e < 16 * 128; e += 256) {
    int r = e >> 7, c = e & 127;
    dstLds[r * LDS_STRIDE + c] = src[r * 128 + c];
  }
#endif
}

__device__ __forceinline__ void stage_wait() {
#if USE_TDM
  if (threadIdx.x < 32) __builtin_amdgcn_s_wait_tensorcnt(0);
#endif
  __syncthreads();
}

__global__ void init_out_kernel(float* out) {
  if (threadIdx.x == 0 && blockIdx.x == 0) out[0] = 0.0f;
}

// a[k] = ||all_x[k,:]||^2 (k<8192), b[i] = ||sub_x[i,:]||^2.
__global__ void __launch_bounds__(128)
norms_kernel(const float* __restrict__ allx, const float* __restrict__ subx,
             float* __restrict__ aNorm, float* __restrict__ bNorm) {
  const int lane = threadIdx.x & 31;
  const int row  = blockIdx.x * 4 + (threadIdx.x >> 5);
  const float* src;
  float* dst;
  if (row < K_DIM) { src = allx + (size_t)row * D_DIM;           dst = aNorm + row; }
  else             { src = subx + (size_t)(row - K_DIM) * D_DIM; dst = bNorm + (row - K_DIM); }
  float acc = 0.0f;
  #pragma unroll
  for (int t = 0; t < 4; ++t) {
    float x = src[lane + t * 32];
    acc += x * x;
  }
  acc = wave_sum32(acc);
  if (lane == 0) *dst = acc;
}

// 8 waves/block, one 16-row strip each; B tiles shared via LDS (TDM staged,
// double buffered). Two WMMA accumulator chains for ILP.
__global__ void __launch_bounds__(256)
contrastive_kernel(const float* __restrict__ hi, const float* __restrict__ hj,
                   float* __restrict__ out) {
  __shared__ float sB[2][16 * LDS_STRIDE];
  const int lane = threadIdx.x & 31;
  const int wv   = threadIdx.x >> 5;                 // 0..7
  const int rowTile = blockIdx.x * 8 + wv;           // 0..511
  const int row0 = rowTile * 16;
  const int l15  = lane & 15;
  const int hh   = lane >> 4;

  const int ra = row0 + l15;
  const float* rowA = (ra < B_ROWS) ? (hi + (size_t)ra * D_DIM)
                                    : (hj + (size_t)(ra - B_ROWS) * D_DIM);
  v2f A[32];
  #pragma unroll
  for (int k = 0; k < 32; ++k)
    A[k] = *(const v2f*)(rowA + 4 * k + 2 * hh);

  float m[8], s[8], pos[8];
  #pragma unroll
  for (int v = 0; v < 8; ++v) { m[v] = -3.0e38f; s[v] = 0.0f; pos[v] = 0.0f; }

  auto tilePtr = [&](int jt) -> const float* {
    int c0 = jt * 16;
    return (c0 < B_ROWS) ? (hi + (size_t)c0 * D_DIM)
                         : (hj + (size_t)(c0 - B_ROWS) * D_DIM);
  };

  stage_issue(sB[0], tilePtr(0));
  stage_wait();

  for (int jt = 0; jt < N_TOT / 16; ++jt) {
    const int cur = jt & 1;
    if (jt + 1 < N_TOT / 16) stage_issue(sB[cur ^ 1], tilePtr(jt + 1));

    const float* tb = sB[cur] + l15 * LDS_STRIDE + 2 * hh;
    v8f C0 = {}, C1 = {};
    #pragma unroll
    for (int k = 0; k < 32; k += 2) {               // 2 independent chains
      v2f Bf0 = *(const v2f*)(tb + 4 * k);
      v2f Bf1 = *(const v2f*)(tb + 4 * k + 4);
      C0 = __builtin_amdgcn_wmma_f32_16x16x4_f32(false, A[k], false, Bf0,
                                                 (short)0, C0, false, false);
      C1 = __builtin_amdgcn_wmma_f32_16x16x4_f32(false, A[k + 1], false, Bf1,
                                                 (short)0, C1, false, false);
    }

    const int c = jt * 16 + l15;                    // this lane's global column
    #pragma unroll
    for (int v = 0; v < 8; ++v) {
      const int r = row0 + v + 8 * hh;              // this element's global row
      float x = INV_T * (C0[v] + C1[v]);
      if (c == ((r + B_ROWS) & (N_TOT - 1))) pos[v] += x;   // positive pair
      if (c == r) x = -INFINITY;                            // mask diagonal
      float mn = fmaxf(m[v], x);
      s[v] = s[v] * __expf(m[v] - mn) + __expf(x - mn);
      m[v] = mn;
    }
    stage_wait();   // TDM done for next tile + protect buffer reuse
  }

  // Merge (m,s,pos) across the 16-lane half-group (offsets 1,2,4,8 stay in-half).
  float local = 0.0f;
  #pragma unroll
  for (int v = 0; v < 8; ++v) {
    float mv = m[v], sv = s[v], pv = pos[v];
    #pragma unroll
    for (int off = 1; off < 16; off <<= 1) {
      float mo = __shfl_xor(mv, off, 32);
      float so = __shfl_xor(sv, off, 32);
      pv += __shfl_xor(pv, off, 32);
      float mn = fmaxf(mv, mo);
      sv = sv * __expf(mv - mn) + so * __expf(mo - mn);
      mv = mn;
    }
    local += (mv + __logf(sv) - pv) * (1.0f / 16.0f);
  }
  local = wave_sum32(local);
  if (lane == 0) atomicAdd(out, local * (1.0f / (float)N_TOT));
}

// 4 waves/block share one K chunk (identical all_x/aNorm addresses -> WGP$ hits)
// across 4 different 16-row strips of G. G streamed once from HBM w/ prefetch.
__global__ void __launch_bounds__(128)
graph_kernel(const float* __restrict__ G, const float* __restrict__ subx,
             const float* __restrict__ allx, const float* __restrict__ aNorm,
             const float* __restrict__ bNorm, float* __restrict__ out) {
  const int lane = threadIdx.x & 31;
  const int wv   = threadIdx.x >> 5;
  const int kch  = blockIdx.x >> 6;                       // 0..7 (shared in block)
  const int tile = (blockIdx.x & 63) * 4 + wv;            // 0..255
  const int i0   = tile * 16;
  const int l15  = lane & 15;
  const int hh   = lane >> 4;
  const int kBeg = kch * (K_DIM / 8);
  const int kEnd = kBeg + (K_DIM / 8);

  const float* rowG = G + (size_t)(i0 + l15) * K_DIM;

  v8f C[8] = {};
  float rAcc = 0.0f, gaAcc = 0.0f;

  for (int k4 = kBeg; k4 < kEnd; k4 += 4) {
    if ((k4 & 63) == 0 && k4 + 256 < K_DIM)
      __builtin_prefetch(rowG + k4 + 256, 0, 0);          // global_prefetch 1KB ahead
    v2f A  = *(const v2f*)(rowG + k4 + 2 * hh);
    v2f a2 = *(const v2f*)(aNorm + k4 + 2 * hh);
    rAcc  += A.x + A.y;                                   // t1 partial
    gaAcc += A.x * a2.x + A.y * a2.y;                     // t2 partial
    const float* bx = allx + (size_t)(k4 + 2 * hh) * D_DIM + l15;
    #pragma unroll
    for (int n = 0; n < 8; ++n) {                         // 8 independent chains
      v2f Bf;
      Bf.x = bx[n * 16];
      Bf.y = bx[n * 16 + D_DIM];
      C[n] = __builtin_amdgcn_wmma_f32_16x16x4_f32(false, A, false, Bf,
                                                   (short)0, C[n], false, false);
    }
  }

  float t3 = 0.0f;
  #pragma unroll
  for (int n = 0; n < 8; ++n) {
    #pragma unroll
    for (int v = 0; v < 8; ++v) {
      const int r = i0 + v + 8 * hh;
      t3 += C[n][v] * subx[(size_t)r * D_DIM + n * 16 + l15];
    }
  }
  float local = bNorm[i0 + l15] * rAcc + gaAcc - 2.0f * t3;
  local = wave_sum32(local);
  if (lane == 0)
    atomicAdd(out, local * (1.0f / ((float)B_ROWS * (float)K_DIM)));
}

extern "C" void kernel_launch(void* const* d_in, const int* in_sizes, int n_in,
                              void* d_out, int out_size, void* d_ws, size_t ws_size,
                              hipStream_t stream) {
  const float* hi = (const float*)d_in[0];   // h_i      [4096,128]
  const float* hj = (const float*)d_in[1];   // h_j      [4096,128]
  const float* G  = (const float*)d_in[2];   // sub_graph[4096,8192]
  const float* sx = (const float*)d_in[3];   // sub_x    [4096,128]
  const float* ax = (const float*)d_in[4];   // all_x    [8192,128]
  float* out = (float*)d_out;
  float* aN = (float*)d_ws;                  // 8192 floats
  float* bN = aN + K_DIM;                    // 4096 floats

  hipLaunchKernelGGL(init_out_kernel, dim3(1), dim3(32), 0, stream, out);
  hipLaunchKernelGGL(norms_kernel, dim3((K_DIM + B_ROWS) / 4), dim3(128), 0, stream,
                     ax, sx, aN, bN);
  hipLaunchKernelGGL(contrastive_kernel, dim3((N_TOT / 16) / 8), dim3(256), 0, stream,
                     hi, hj, out);
  hipLaunchKernelGGL(graph_kernel, dim3(512), dim3(128), 0, stream,
                     G, sx, ax, aN, bN, out);
}